// MGraphModel_76888504533069
// MI455X (gfx1250) — compile-verified
//
#include <hip/hip_runtime.h>
#include <hip/hip_bf16.h>
#include <math.h>

// ---------------------------------------------------------------------------
// MI455X (gfx1250). All GEMMs on V_WMMA_F32_16X16X4_F32 (fp32 matrix pipes:
// the model is bandwidth/latency bound, so low-precision WMMA buys nothing).
// Wave32: one wave per 16x16 C tile. GEMM inner loops are branch-free:
// operands are padded/aligned so no predication is needed inside the K loop.
// ---------------------------------------------------------------------------

typedef __attribute__((ext_vector_type(2))) float v2f;
typedef __attribute__((ext_vector_type(8))) float v8f;

#define CDIV(a, b) (((a) + (b) - 1) / (b))

constexpr int BATCH     = 32;
constexpr int SEQ       = 100;
constexpr int SEQ_PAD   = 112;                                // 7 * 16
constexpr int FIN       = 65;
constexpr int FIN_PAD   = 68;                                 // mult of 4, even
constexpr int NHEAD_GAT = 4;
constexpr int FOUT      = 64;
constexpr int DMODEL    = 256;
constexpr int D_STATE   = 64;
constexpr int D_CONV    = 4;
constexpr int HEADDIM   = 64;
constexpr int D_INNER   = 1024;                               // 4 * 256
constexpr int NHEADS    = 16;                                 // 1024 / 64
constexpr int CONV_DIM  = D_INNER + 2 * D_STATE;              // 1152
constexpr int D_IN_PROJ = 2 * D_INNER + 2 * D_STATE + NHEADS; // 2192
constexpr int BIAS_DIM  = 32;
constexpr int HID1      = 1024;
constexpr int HID2      = 128;
constexpr int TOK       = BATCH * SEQ;                        // 3200
constexpr int KFLAT     = SEQ * DMODEL;                       // 25600
constexpr int KSPLIT    = 16;
constexpr int KCHUNK    = KFLAT / KSPLIT;                     // 1600

// ---------------------------------------------------------------------------
// Branch-free WMMA K-loop. Layout (fp32 16x16x4):
//   A (16x4): lanes 0-15 hold K={0,1}, lanes 16-31 K={2,3}; row = lane&15.
//   B (4x16): col = lane&15, same K split. C/D: VGPR v -> row v + 8*(lane>>4).
// Arow must be 8B aligned (lda even everywhere) -> A is one v2f load/step.
// ---------------------------------------------------------------------------
template <int K>
__device__ inline v8f wmma_kloop(const float* __restrict__ Arow,
                                 const float* __restrict__ Bcol, int ldb,
                                 v8f acc) {
#pragma unroll 4
  for (int k0 = 0; k0 < K; k0 += 4) {
    const v2f a = *(const v2f*)(Arow + k0);
    v2f b;
    b.x = Bcol[(size_t)k0 * ldb];
    b.y = Bcol[(size_t)k0 * ldb + ldb];
    acc = __builtin_amdgcn_wmma_f32_16x16x4_f32(false, a, false, b,
                                                (short)0, acc, false, false);
  }
  return acc;
}

// Generic aligned GEMM: C[M,Nn] = A[M,K] @ B[K,Nn]; M,Nn mult of 16, K mult 4.
template <int K>
__global__ void k_gemm_aligned(const float* __restrict__ A, int lda,
                               const float* __restrict__ Bm, int ldb,
                               float* __restrict__ C, int ldc) {
  const int nt = blockIdx.x, mt = blockIdx.y;
  const int lane = threadIdx.x & 31, half = lane >> 4, idx = lane & 15;
  const float* Arow = A + (size_t)(mt * 16 + idx) * lda + half * 2;
  const float* Bcol = Bm + (size_t)(half * 2) * ldb + nt * 16 + idx;
  v8f acc = {0.f, 0.f, 0.f, 0.f, 0.f, 0.f, 0.f, 0.f};
  acc = wmma_kloop<K>(Arow, Bcol, ldb, acc);
  float* Cp = C + (size_t)(mt * 16) * ldc + nt * 16 + idx;
#pragma unroll
  for (int v = 0; v < 8; ++v) Cp[(size_t)(half * 8 + v) * ldc] = acc[v];
}

// Deterministic split-K GEMM (per-chunk partials, fixed-order reduce; no FP
// atomics so graph replays are bit-identical).
__global__ void k_gemm_splitk(const float* __restrict__ A, int lda,
                              const float* __restrict__ Bm, int ldb,
                              float* __restrict__ Cpart, int M, int Nn) {
  const int nt = blockIdx.x, mt = blockIdx.y, kc = blockIdx.z;
  const int lane = threadIdx.x & 31, half = lane >> 4, idx = lane & 15;
  const int kbeg = kc * KCHUNK;
  const float* Arow = A + (size_t)(mt * 16 + idx) * lda + half * 2 + kbeg;
  const float* Bcol = Bm + (size_t)(half * 2 + kbeg) * ldb + nt * 16 + idx;
  v8f acc = {0.f, 0.f, 0.f, 0.f, 0.f, 0.f, 0.f, 0.f};
  acc = wmma_kloop<KCHUNK>(Arow, Bcol, ldb, acc);
  float* Cp = Cpart + (size_t)kc * M * Nn + (size_t)(mt * 16) * Nn + nt * 16 + idx;
#pragma unroll
  for (int v = 0; v < 8; ++v) Cp[(size_t)(half * 8 + v) * Nn] = acc[v];
}

__global__ void k_reduce_parts(const float* __restrict__ part, float* __restrict__ out,
                               int n, int nparts) {
  const int i = blockIdx.x * blockDim.x + threadIdx.x;
  if (i >= n) return;
  float s = 0.f;
  for (int p = 0; p < nparts; ++p) s += part[(size_t)p * n + i];
  out[i] = s;
}

// ---------------------------------------------------------------------------
// GAT stage. X and gat_W are zero-padded once so the WMMA loops are clean.
// ---------------------------------------------------------------------------
__global__ void k_pad_x(const float* __restrict__ X, float* __restrict__ Xp) {
  const int i = blockIdx.x * blockDim.x + threadIdx.x; // BATCH*SEQ_PAD*FIN_PAD
  if (i >= BATCH * SEQ_PAD * FIN_PAD) return;
  const int f = i % FIN_PAD, n = (i / FIN_PAD) % SEQ_PAD, b = i / (FIN_PAD * SEQ_PAD);
  Xp[i] = (n < SEQ && f < FIN) ? X[((size_t)b * SEQ + n) * FIN + f] : 0.f;
}

__global__ void k_pad_w(const float* __restrict__ W, float* __restrict__ Wp) {
  const int i = blockIdx.x * blockDim.x + threadIdx.x; // H*FIN_PAD*FOUT
  if (i >= NHEAD_GAT * FIN_PAD * FOUT) return;
  const int o = i % FOUT, f = (i / FOUT) % FIN_PAD, h = i / (FOUT * FIN_PAD);
  Wp[i] = (f < FIN) ? W[((size_t)h * FIN + f) * FOUT + o] : 0.f;
}

// Wh[bh][0:112][0:64] = Xp[b] (112x68) @ Wp[h] (68x64); pad rows of Xp are 0.
__global__ void k_gat_wh(const float* __restrict__ Xp, const float* __restrict__ Wp,
                         float* __restrict__ Wh) {
  const int bh = blockIdx.y, b = bh / NHEAD_GAT, h = bh % NHEAD_GAT;
  const int mt = blockIdx.x / 4, nt = blockIdx.x % 4;
  const int lane = threadIdx.x & 31, half = lane >> 4, idx = lane & 15;
  const float* Arow = Xp + (size_t)b * SEQ_PAD * FIN_PAD
                         + (size_t)(mt * 16 + idx) * FIN_PAD + half * 2;
  const float* Bcol = Wp + (size_t)h * FIN_PAD * FOUT
                         + (size_t)(half * 2) * FOUT + nt * 16 + idx;
  v8f acc = {0.f, 0.f, 0.f, 0.f, 0.f, 0.f, 0.f, 0.f};
  acc = wmma_kloop<FIN_PAD>(Arow, Bcol, FOUT, acc);
  float* Cp = Wh + (size_t)bh * SEQ_PAD * FOUT + (size_t)(mt * 16) * FOUT + nt * 16 + idx;
#pragma unroll
  for (int v = 0; v < 8; ++v) Cp[(size_t)(half * 8 + v) * FOUT] = acc[v];
}

__global__ void k_gat_f(const float* __restrict__ Wh, const float* __restrict__ a1,
                        const float* __restrict__ a2, float* __restrict__ f1,
                        float* __restrict__ f2) {
  const int i = blockIdx.x * blockDim.x + threadIdx.x; // over B*H*SEQ
  if (i >= BATCH * NHEAD_GAT * SEQ) return;
  const int n = i % SEQ, bh = i / SEQ, h = bh % NHEAD_GAT;
  const float* w  = Wh + ((size_t)bh * SEQ_PAD + n) * FOUT;
  const float* v1 = a1 + h * FOUT;
  const float* v2 = a2 + h * FOUT;
  float s1 = 0.f, s2 = 0.f;
  for (int o = 0; o < FOUT; ++o) { const float x = w[o]; s1 += x * v1[o]; s2 += x * v2[o]; }
  f1[i] = s1; f2[i] = s2;
}

// att rows live in a SEQ_PAD-row buffer; pad rows are never written (their
// WMMA products land in discarded C rows only — rows are independent).
__global__ void k_gat_att(const float* __restrict__ f1, const float* __restrict__ f2,
                          const float* __restrict__ Aadj, float* __restrict__ att) {
  const int row = blockIdx.x;            // b*H*SEQ
  const int n = row % SEQ, bh = row / SEQ, b = bh / NHEAD_GAT;
  __shared__ float red[128];
  const float f1v = f1[row];
  const float* f2row = f2 + (size_t)bh * SEQ;
  const float* arow  = Aadj + ((size_t)b * SEQ + n) * SEQ;
  const int m = threadIdx.x;
  const bool valid = (m < SEQ);
  float e = -3.0e38f;
  if (valid) {
    const float x  = f1v + f2row[m];
    const float lr = (x > 0.f) ? x : 0.2f * x;
    e = (arow[m] > 0.f) ? lr : -9.0e15f;
  }
  red[threadIdx.x] = e;
  __syncthreads();
  for (int s = 64; s > 0; s >>= 1) {
    if (threadIdx.x < s) red[threadIdx.x] = fmaxf(red[threadIdx.x], red[threadIdx.x + s]);
    __syncthreads();
  }
  const float mx = red[0];
  __syncthreads();
  const float ex = valid ? __expf(e - mx) : 0.f;
  red[threadIdx.x] = ex;
  __syncthreads();
  for (int s = 64; s > 0; s >>= 1) {
    if (threadIdx.x < s) red[threadIdx.x] += red[threadIdx.x + s];
    __syncthreads();
  }
  if (valid) att[((size_t)bh * SEQ_PAD + n) * SEQ + m] = ex / red[0];
}

// hout = elu(att @ Wh) stored transposed into ax[b][n][h*FOUT+o].
__global__ void k_gat_out(const float* __restrict__ att, const float* __restrict__ Wh,
                          float* __restrict__ ax) {
  const int bh = blockIdx.y, b = bh / NHEAD_GAT, h = bh % NHEAD_GAT;
  const int mt = blockIdx.x / 4, nt = blockIdx.x % 4;
  const int lane = threadIdx.x & 31, half = lane >> 4, idx = lane & 15;
  const float* Arow = att + (size_t)bh * SEQ_PAD * SEQ
                          + (size_t)(mt * 16 + idx) * SEQ + half * 2;
  const float* Bcol = Wh + (size_t)bh * SEQ_PAD * FOUT
                         + (size_t)(half * 2) * FOUT + nt * 16 + idx;
  v8f acc = {0.f, 0.f, 0.f, 0.f, 0.f, 0.f, 0.f, 0.f};
  acc = wmma_kloop<SEQ>(Arow, Bcol, FOUT, acc);   // K = 100, mult of 4
  const int o = nt * 16 + idx;
#pragma unroll
  for (int v = 0; v < 8; ++v) {
    const int n = mt * 16 + half * 8 + v;
    if (n < SEQ) {
      const float x = acc[v];
      ax[((size_t)b * SEQ + n) * DMODEL + h * FOUT + o] =
          (x > 0.f) ? x : (__expf(x) - 1.f);
    }
  }
}

// ---------------------------------------------------------------------------
// Structural bias + gate
// ---------------------------------------------------------------------------
__global__ void k_bias(const float* __restrict__ Aadj, const float* __restrict__ spd,
                       const float* __restrict__ bond, const float* __restrict__ dege,
                       float* __restrict__ bias_all) {
  const int bn = blockIdx.x, b = bn / SEQ, n = bn % SEQ;
  const int j = threadIdx.x;               // 32 = BIAS_DIM
  const float* arow = Aadj + ((size_t)b * SEQ + n) * SEQ;
  float s_spd = 0.f, s_bond = 0.f, degsum = 0.f;
  for (int m = 0; m < SEQ; ++m) {
    const float av = arow[m];
    const bool adj = (av > 0.f);
    const int dist = (m == n) ? 0 : (adj ? 1 : 2);
    s_spd  += spd[dist * BIAS_DIM + j];
    s_bond += bond[(adj ? 1 : 0) * BIAS_DIM + j];
    degsum += av;
  }
  const int dg = (int)fminf(degsum, 15.0f);
  bias_all[(size_t)bn * BIAS_DIM + j] =
      (s_spd + s_bond) * (1.0f / SEQ) + dege[dg * BIAS_DIM + j];
}

__global__ void k_gate(const float* __restrict__ bias_all, const float* __restrict__ gateW,
                       const float* __restrict__ gateB, const float* __restrict__ biasP,
                       float* __restrict__ ax) {
  const int i = blockIdx.x * blockDim.x + threadIdx.x; // B*SEQ*D
  if (i >= TOK * DMODEL) return;
  const int d = i % DMODEL, bn = i / DMODEL;
  const float* bb = bias_all + (size_t)bn * BIAS_DIM;
  float g = gateB[d], pp = 0.f;
  for (int j = 0; j < BIAS_DIM; ++j) {
    const float bj = bb[j];
    g  += bj * gateW[j * DMODEL + d];
    pp += bj * biasP[j * DMODEL + d];
  }
  ax[i] += pp / (1.f + __expf(-g));
}

// ---------------------------------------------------------------------------
// Mamba block pieces
// ---------------------------------------------------------------------------
__global__ void k_conv(const float* __restrict__ zxbcdt, const float* __restrict__ convW,
                       const float* __restrict__ convB, float* __restrict__ xBC) {
  const int i = blockIdx.x * blockDim.x + threadIdx.x; // B*SEQ*CONV_DIM
  if (i >= TOK * CONV_DIM) return;
  const int c = i % CONV_DIM, t = (i / CONV_DIM) % SEQ, b = i / (CONV_DIM * SEQ);
  float s = convB[c];
#pragma unroll
  for (int k = 0; k < D_CONV; ++k) {
    const int tt = t + k - (D_CONV - 1);
    if (tt >= 0)
      s += zxbcdt[((size_t)b * SEQ + tt) * D_IN_PROJ + D_INNER + c] * convW[c * D_CONV + k];
  }
  xBC[i] = s / (1.f + __expf(-s));   // silu
}

__global__ void k_dt(const float* __restrict__ zxbcdt, const float* __restrict__ dt_bias,
                     const float* __restrict__ A_log, float* __restrict__ dt,
                     float* __restrict__ dA) {
  const int i = blockIdx.x * blockDim.x + threadIdx.x; // B*SEQ*NHEADS
  if (i >= TOK * NHEADS) return;
  const int h = i % NHEADS;
  const float x = zxbcdt[(size_t)(i / NHEADS) * D_IN_PROJ + D_INNER + CONV_DIM + h] + dt_bias[h];
  const float sp = (x > 20.f) ? x : log1pf(__expf(x));   // softplus
  dt[i] = sp;
  dA[i] = __expf(sp * -__expf(A_log[h]));
}

// Selective scan: one block per (batch, head); 256 threads hold the 64x64
// state in registers (16 floats each: fixed p = tid/4, n-range of 16).
__global__ void k_scan(const float* __restrict__ xBC, const float* __restrict__ dt,
                       const float* __restrict__ dA, const float* __restrict__ Dp,
                       float* __restrict__ y) {
  const int bh = blockIdx.x, b = bh / NHEADS, hd = bh % NHEADS;
  const int tid = threadIdx.x, p = tid >> 2, sub = tid & 3;
  float hstate[16];
#pragma unroll
  for (int q = 0; q < 16; ++q) hstate[q] = 0.f;
  __shared__ float sx[HEADDIM];
  __shared__ float sB[D_STATE];
  __shared__ float sC[D_STATE];
  __shared__ float sy[HEADDIM * 4];
  const float Dval = Dp[hd];
  for (int t = 0; t < SEQ; ++t) {
    const float* xrow = xBC + ((size_t)b * SEQ + t) * CONV_DIM;
    if (tid < HEADDIM)                  sx[tid] = xrow[hd * HEADDIM + tid];
    else if (tid < HEADDIM + D_STATE)   sB[tid - HEADDIM] = xrow[D_INNER + (tid - HEADDIM)];
    else if (tid < HEADDIM + 2*D_STATE) sC[tid - HEADDIM - D_STATE] =
                                            xrow[D_INNER + D_STATE + (tid - HEADDIM - D_STATE)];
    __syncthreads();
    const float dA_t = dA[((size_t)b * SEQ + t) * NHEADS + hd];
    const float dt_t = dt[((size_t)b * SEQ + t) * NHEADS + hd];
    const float xv = sx[p] * dt_t;
    float acc = 0.f;
#pragma unroll
    for (int q = 0; q < 16; ++q) {
      const int nn = sub * 16 + q;
      const float hnew = hstate[q] * dA_t + xv * sB[nn];
      hstate[q] = hnew;
      acc += hnew * sC[nn];
    }
    sy[p * 4 + sub] = acc;
    __syncthreads();
    if (sub == 0) {
      const float tot = sy[p * 4] + sy[p * 4 + 1] + sy[p * 4 + 2] + sy[p * 4 + 3];
      y[((size_t)b * SEQ + t) * D_INNER + hd * HEADDIM + p] = tot + Dval * sx[p];
    }
    __syncthreads();
  }
}

// y = y*silu(z); y *= rsqrt(mean(y^2)+eps) * norm_w   (in place ok)
__global__ void k_rmsnorm(const float* __restrict__ yin, const float* __restrict__ zxbcdt,
                          const float* __restrict__ norm_w, float* __restrict__ yout) {
  const int bn = blockIdx.x;
  const float* yr = yin + (size_t)bn * D_INNER;
  const float* zr = zxbcdt + (size_t)bn * D_IN_PROJ; // z = first D_INNER cols
  __shared__ float red[256];
  float vals[4], ss = 0.f;
#pragma unroll
  for (int q = 0; q < 4; ++q) {
    const int idx = threadIdx.x + q * 256;
    const float z = zr[idx];
    const float v = yr[idx] * (z / (1.f + __expf(-z)));
    vals[q] = v; ss += v * v;
  }
  red[threadIdx.x] = ss;
  __syncthreads();
  for (int s = 128; s > 0; s >>= 1) {
    if (threadIdx.x < s) red[threadIdx.x] += red[threadIdx.x + s];
    __syncthreads();
  }
  const float scale = rsqrtf(red[0] / D_INNER + 1e-5f);
#pragma unroll
  for (int q = 0; q < 4; ++q) {
    const int idx = threadIdx.x + q * 256;
    yout[(size_t)bn * D_INNER + idx] = vals[q] * scale * norm_w[idx];
  }
}

// ---------------------------------------------------------------------------
// LayerNorm over DMODEL, batch-norm+relu over rows
// ---------------------------------------------------------------------------
__global__ void k_layernorm(const float* __restrict__ xin, const float* __restrict__ g,
                            const float* __restrict__ bta, float* __restrict__ xout) {
  const int bn = blockIdx.x, tid = threadIdx.x;   // 256 = DMODEL
  const float v = xin[(size_t)bn * DMODEL + tid];
  __shared__ float r1[256], r2[256];
  r1[tid] = v; r2[tid] = v * v;
  __syncthreads();
  for (int s = 128; s > 0; s >>= 1) {
    if (tid < s) { r1[tid] += r1[tid + s]; r2[tid] += r2[tid + s]; }
    __syncthreads();
  }
  const float mu  = r1[0] / DMODEL;
  const float var = r2[0] / DMODEL - mu * mu;
  xout[(size_t)bn * DMODEL + tid] = (v - mu) * rsqrtf(var + 1e-5f) * g[tid] + bta[tid];
}

__global__ void k_bnrelu(const float* __restrict__ x, const float* __restrict__ bias,
                         const float* __restrict__ g, const float* __restrict__ bb,
                         float* __restrict__ out, int cols) {
  const int j = blockIdx.x, r = threadIdx.x;      // 32 rows
  const float v = x[(size_t)r * cols + j] + bias[j];
  __shared__ float r1[32], r2[32];
  r1[r] = v; r2[r] = v * v;
  __syncthreads();
  for (int s = 16; s > 0; s >>= 1) {
    if (r < s) { r1[r] += r1[r + s]; r2[r] += r2[r + s]; }
    __syncthreads();
  }
  const float mu  = r1[0] / BATCH;
  const float var = r2[0] / BATCH - mu * mu;
  const float o = (v - mu) * rsqrtf(var + 1e-5f) * g[j] + bb[j];
  out[(size_t)r * cols + j] = fmaxf(o, 0.f);
}

// ---------------------------------------------------------------------------
// Host orchestration
// ---------------------------------------------------------------------------
static void mamba_block(const float* ax_in, float* ax_out, void* const* d_in, int base,
                        float* zxbcdt, float* xBC, float* dtb, float* dAb, float* yb,
                        hipStream_t stream) {
  const float* in_proj  = (const float*)d_in[base + 0];
  const float* conv_w   = (const float*)d_in[base + 1];
  const float* conv_b   = (const float*)d_in[base + 2];
  const float* dt_bias  = (const float*)d_in[base + 3];
  const float* A_log    = (const float*)d_in[base + 4];
  const float* Dp       = (const float*)d_in[base + 5];
  const float* norm_w   = (const float*)d_in[base + 6];
  const float* out_proj = (const float*)d_in[base + 7];

  // in_proj: (3200 x 2192) = (3200 x 256) @ (256 x 2192)
  k_gemm_aligned<DMODEL><<<dim3(D_IN_PROJ / 16, TOK / 16), 32, 0, stream>>>(
      ax_in, DMODEL, in_proj, D_IN_PROJ, zxbcdt, D_IN_PROJ);
  k_conv<<<CDIV(TOK * CONV_DIM, 256), 256, 0, stream>>>(zxbcdt, conv_w, conv_b, xBC);
  k_dt<<<CDIV(TOK * NHEADS, 256), 256, 0, stream>>>(zxbcdt, dt_bias, A_log, dtb, dAb);
  k_scan<<<BATCH * NHEADS, 256, 0, stream>>>(xBC, dtb, dAb, Dp, yb);
  k_rmsnorm<<<TOK, 256, 0, stream>>>(yb, zxbcdt, norm_w, yb);
  // out_proj: (3200 x 256) = (3200 x 1024) @ (1024 x 256)
  k_gemm_aligned<D_INNER><<<dim3(DMODEL / 16, TOK / 16), 32, 0, stream>>>(
      yb, D_INNER, out_proj, DMODEL, ax_out, DMODEL);
}

extern "C" void kernel_launch(void* const* d_in, const int* in_sizes, int n_in,
                              void* d_out, int out_size, void* d_ws, size_t ws_size,
                              hipStream_t stream) {
  (void)in_sizes; (void)n_in; (void)out_size; (void)ws_size;
  const float* X    = (const float*)d_in[0];
  const float* Aadj = (const float*)d_in[1];
  const float* gatW = (const float*)d_in[2];
  const float* a1   = (const float*)d_in[3];
  const float* a2   = (const float*)d_in[4];
  const float* spd  = (const float*)d_in[5];
  const float* bond = (const float*)d_in[6];
  const float* dege = (const float*)d_in[7];
  const float* bprj = (const float*)d_in[8];
  const float* gW   = (const float*)d_in[9];
  const float* gB   = (const float*)d_in[10];
  // 11..18 = mamba0, 19..26 = mamba1
  const float* ln_g = (const float*)d_in[27];
  const float* ln_b = (const float*)d_in[28];
  const float* W1   = (const float*)d_in[29];
  const float* b1   = (const float*)d_in[30];
  const float* bn1g = (const float*)d_in[31];
  const float* bn1b = (const float*)d_in[32];
  const float* W2   = (const float*)d_in[33];
  const float* b2   = (const float*)d_in[34];
  const float* bn2g = (const float*)d_in[35];
  const float* bn2b = (const float*)d_in[36];
  float* out = (float*)d_out;

  float* ws = (float*)d_ws;
  size_t off = 0;
  auto alloc = [&](size_t n) { float* p = ws + off; off += (n + 1) & ~(size_t)1; return p; };
  float* Xp       = alloc((size_t)BATCH * SEQ_PAD * FIN_PAD);            // 243712
  float* Wp       = alloc((size_t)NHEAD_GAT * FIN_PAD * FOUT);           // 17408
  float* Wh       = alloc((size_t)BATCH * NHEAD_GAT * SEQ_PAD * FOUT);   // 917504
  float* f1       = alloc((size_t)BATCH * NHEAD_GAT * SEQ);
  float* f2       = alloc((size_t)BATCH * NHEAD_GAT * SEQ);
  float* att      = alloc((size_t)BATCH * NHEAD_GAT * SEQ_PAD * SEQ);    // 1.43M
  float* bias_all = alloc((size_t)TOK * BIAS_DIM);
  float* axA      = alloc((size_t)TOK * DMODEL);
  float* axB      = alloc((size_t)TOK * DMODEL);
  float* zxbcdt   = alloc((size_t)TOK * D_IN_PROJ);                      // 7.0M
  float* xBC      = alloc((size_t)TOK * CONV_DIM);                       // 3.7M
  float* dtb      = alloc((size_t)TOK * NHEADS);
  float* dAb      = alloc((size_t)TOK * NHEADS);
  float* yb       = alloc((size_t)TOK * D_INNER);                        // 3.3M
  float* part1    = alloc((size_t)KSPLIT * BATCH * HID1);                // 524288
  float* out1     = alloc((size_t)BATCH * HID1);
  float* out1b    = alloc((size_t)BATCH * HID1);
  float* out2     = alloc((size_t)BATCH * HID2);

  // ---- GAT ----
  k_pad_x<<<CDIV(BATCH * SEQ_PAD * FIN_PAD, 256), 256, 0, stream>>>(X, Xp);
  k_pad_w<<<CDIV(NHEAD_GAT * FIN_PAD * FOUT, 256), 256, 0, stream>>>(gatW, Wp);
  k_gat_wh<<<dim3(28, BATCH * NHEAD_GAT), 32, 0, stream>>>(Xp, Wp, Wh);
  k_gat_f<<<CDIV(BATCH * NHEAD_GAT * SEQ, 256), 256, 0, stream>>>(Wh, a1, a2, f1, f2);
  k_gat_att<<<BATCH * NHEAD_GAT * SEQ, 128, 0, stream>>>(f1, f2, Aadj, att);
  k_gat_out<<<dim3(28, BATCH * NHEAD_GAT), 32, 0, stream>>>(att, Wh, axA);

  // ---- structural bias + gate ----
  k_bias<<<TOK, BIAS_DIM, 0, stream>>>(Aadj, spd, bond, dege, bias_all);
  k_gate<<<CDIV(TOK * DMODEL, 256), 256, 0, stream>>>(bias_all, gW, gB, bprj, axA);

  // ---- Mamba x2 ----
  mamba_block(axA, axB, d_in, 11, zxbcdt, xBC, dtb, dAb, yb, stream);
  mamba_block(axB, axA, d_in, 19, zxbcdt, xBC, dtb, dAb, yb, stream);

  // ---- LayerNorm -> (32, 25600) flat ----
  k_layernorm<<<TOK, DMODEL, 0, stream>>>(axA, ln_g, ln_b, axB);

  // ---- proj1: (32,25600)@(25600,1024), deterministic split-K ----
  k_gemm_splitk<<<dim3(HID1 / 16, BATCH / 16, KSPLIT), 32, 0, stream>>>(
      axB, KFLAT, W1, HID1, part1, BATCH, HID1);
  k_reduce_parts<<<CDIV(BATCH * HID1, 256), 256, 0, stream>>>(
      part1, out1, BATCH * HID1, KSPLIT);
  k_bnrelu<<<HID1, BATCH, 0, stream>>>(out1, b1, bn1g, bn1b, out1b, HID1);

  // ---- proj2: (32,1024)@(1024,128) ----
  k_gemm_aligned<HID1><<<dim3(HID2 / 16, BATCH / 16), 32, 0, stream>>>(
      out1b, HID1, W2, HID2, out2, HID2);
  k_bnrelu<<<HID2, BATCH, 0, stream>>>(out2, b2, bn2g, bn2b, out, HID2);
}